// TaskAlignedAssignerV2_82325933129983
// MI455X (gfx1250) — compile-verified
//
#include <hip/hip_runtime.h>
#include <hip/hip_bf16.h>
#include <stdint.h>

#define TOPK 13
#define EPSV 1e-9f
#define MAX_GRIDS 8448   // grids = 8400 for IMG=640; padded

typedef float v4f __attribute__((ext_vector_type(4)));
typedef float v2f __attribute__((ext_vector_type(2)));

// ---------------------------------------------------------------------------
// K0: zero-fill the (mostly sparse) output with non-temporal 128-bit stores.
// ---------------------------------------------------------------------------
__global__ void zero_out_kernel(float* __restrict__ out, long long n4, long long n) {
    long long i = (long long)blockIdx.x * blockDim.x + threadIdx.x;
    if (i < n4) {
        v4f z = {0.f, 0.f, 0.f, 0.f};
        __builtin_nontemporal_store(z, ((v4f*)out) + i);
    }
    if (i == 0) {
        for (long long j = n4 * 4; j < n; ++j) out[j] = 0.f;
    }
}

// ---------------------------------------------------------------------------
// K1: per-(t,g) metrics: IoU, in-gt mask, score gather.
//   metric  = mask_in ? score * iou^6 : 0
//   ovl_enc = mask_in ? iou + 1      : 0   (sign encodes mask_in)
// ---------------------------------------------------------------------------
__global__ __launch_bounds__(256)
void metrics_kernel(const float* __restrict__ pd_scores,
                    const float* __restrict__ pd_bboxes,
                    const float* __restrict__ anc,
                    const int*   __restrict__ gt_labels,
                    const float* __restrict__ gt_bboxes,
                    float* __restrict__ metric,
                    float* __restrict__ ovl_enc,
                    int grids, int nc) {
    int g = blockIdx.x * blockDim.x + threadIdx.x;
    int t = blockIdx.y;
    if (g >= grids) return;

    size_t base = (size_t)t * grids + g;
    int lab = gt_labels[t];
    const float* sp = pd_scores + base * (size_t)nc + lab;
    __builtin_prefetch(sp, 0, 3);                    // global_prefetch_b8: hide gather latency

    v4f gt = *(const v4f*)(gt_bboxes + (size_t)t * 4);
    v4f pb = *(const v4f*)(pd_bboxes + base * 4);
    v2f a  = *(const v2f*)(anc + (size_t)g * 2);

    float ix1 = fmaxf(gt.x, pb.x), iy1 = fmaxf(gt.y, pb.y);
    float ix2 = fminf(gt.z, pb.z), iy2 = fminf(gt.w, pb.w);
    float inter = fmaxf(ix2 - ix1, 0.f) * fmaxf(iy2 - iy1, 0.f);
    float a1 = fmaxf(gt.z - gt.x, 0.f) * fmaxf(gt.w - gt.y, 0.f);
    float a2 = fmaxf(pb.z - pb.x, 0.f) * fmaxf(pb.w - pb.y, 0.f);
    float iou = inter / (a1 + a2 - inter + EPSV);

    float mind = fminf(fminf(a.x - gt.x, a.y - gt.y),
                       fminf(gt.z - a.x, gt.w - a.y));
    bool in = mind > EPSV;

    float sc  = *sp;
    float i2  = iou * iou;
    float align = sc * (i2 * i2 * i2);

    metric[base]  = in ? align : 0.f;
    ovl_enc[base] = in ? (iou + 1.f) : 0.f;
}

// ---------------------------------------------------------------------------
// K2: per-row exact top-13 with lowest-index tie-break.
// Stages the metric row into LDS via global_load_async_to_lds_b128 (ASYNCcnt),
// then 13 argmax passes re-read LDS instead of HBM.
// ---------------------------------------------------------------------------
__global__ __launch_bounds__(256)
void topk_kernel(const float* __restrict__ metric,
                 const float* __restrict__ ovl_enc,
                 int*   __restrict__ pos_idx,
                 float* __restrict__ pos_am,
                 float* __restrict__ rowscale,
                 int grids) {
    __shared__ float sm[MAX_GRIDS];
    __shared__ float redv[256];
    __shared__ int   redi[256];
    __shared__ int   ch_i[TOPK];
    __shared__ float ch_v[TOPK];
    __shared__ float sc_am[TOPK];
    __shared__ float sc_ov[TOPK];

    int t   = blockIdx.x;
    int tid = threadIdx.x;
    const float* row = metric + (size_t)t * grids;

    // --- async copy row -> LDS (CDNA5 async-tensor path, tracked by ASYNCcnt)
    uint32_t lds_base = (uint32_t)(uintptr_t)(&sm[0]);  // low 32 bits of generic ptr = LDS offset
    int chunks = grids >> 2;                            // 16-byte chunks
    for (int c = tid; c < chunks; c += 256) {
        uint32_t lda = lds_base + (uint32_t)c * 16u;
        uint64_t ga  = (uint64_t)(uintptr_t)(row + (size_t)c * 4);
        asm volatile("global_load_async_to_lds_b128 %0, %1, off"
                     :: "v"(lda), "v"(ga) : "memory");
    }
    for (int i = (chunks << 2) + tid; i < grids; i += 256) {   // scalar tail
        uint32_t lda = lds_base + (uint32_t)i * 4u;
        uint64_t ga  = (uint64_t)(uintptr_t)(row + i);
        asm volatile("global_load_async_to_lds_b32 %0, %1, off"
                     :: "v"(lda), "v"(ga) : "memory");
    }
    asm volatile("s_wait_asynccnt 0x0" ::: "memory");
    __syncthreads();

    // --- 13 iterative argmax passes (exact jax.lax.top_k tie semantics)
    for (int k = 0; k < TOPK; ++k) {
        float bv = -2.f; int bi = 0x7fffffff;
        for (int i = tid; i < grids; i += 256) {
            float v = sm[i];
            if (v > bv) { bv = v; bi = i; }   // increasing i => keeps lowest index on ties
        }
        redv[tid] = bv; redi[tid] = bi;
        __syncthreads();
        for (int s = 128; s > 0; s >>= 1) {
            if (tid < s) {
                float v2 = redv[tid + s]; int j2 = redi[tid + s];
                if (v2 > redv[tid] || (v2 == redv[tid] && j2 < redi[tid])) {
                    redv[tid] = v2; redi[tid] = j2;
                }
            }
            __syncthreads();
        }
        if (tid == 0) {
            ch_i[k] = redi[0];
            ch_v[k] = redv[0];
            sm[redi[0]] = -1.f;   // exclude from later passes (metrics are >= 0)
        }
        __syncthreads();
    }

    // --- positives + per-row normalization scale
    if (tid < TOPK) {
        int   g  = ch_i[tid];
        float mv = ch_v[tid];
        float oe = ovl_enc[(size_t)t * grids + g];
        bool  in = oe > 0.f;                      // mask_in at chosen anchor
        float am = in ? mv : 0.f;                 // align_metric * mask_pos
        pos_idx[t * TOPK + tid] = in ? g : -1;
        pos_am [t * TOPK + tid] = am;
        sc_am[tid] = am;
        sc_ov[tid] = in ? (oe - 1.f) : 0.f;       // overlaps * mask_pos
    }
    __syncthreads();
    if (tid == 0) {
        float pa = 0.f, po = 0.f;
        for (int k = 0; k < TOPK; ++k) {
            pa = fmaxf(pa, sc_am[k]);
            po = fmaxf(po, sc_ov[k]);
        }
        rowscale[t] = po / (pa + EPSV);
    }
}

// ---------------------------------------------------------------------------
// K3: scatter the <=TN*13 positives into the zeroed outputs.
// ---------------------------------------------------------------------------
__global__ void scatter_kernel(const int*   __restrict__ gt_labels,
                               const float* __restrict__ gt_bboxes,
                               const int*   __restrict__ pos_idx,
                               const float* __restrict__ pos_am,
                               const float* __restrict__ rowscale,
                               int*   __restrict__ out_labels,
                               float* __restrict__ out_bboxes,
                               float* __restrict__ out_scores,
                               float* __restrict__ out_mask,
                               int grids, int nc, int tn) {
    int i = blockIdx.x * blockDim.x + threadIdx.x;
    if (i >= tn * TOPK) return;
    int t = i / TOPK;
    int g = pos_idx[i];
    if (g < 0) return;

    size_t base = (size_t)t * grids + g;
    int lab = gt_labels[t];
    out_labels[base] = lab;
    v4f gt = *(const v4f*)(gt_bboxes + (size_t)t * 4);
    *(v4f*)(out_bboxes + base * 4) = gt;
    out_scores[base * (size_t)nc + lab] = pos_am[i] * rowscale[t];
    out_mask[base] = 1.0f;
}

// ---------------------------------------------------------------------------
extern "C" void kernel_launch(void* const* d_in, const int* in_sizes, int n_in,
                              void* d_out, int out_size, void* d_ws, size_t ws_size,
                              hipStream_t stream) {
    const float* pd_scores = (const float*)d_in[0];
    const float* pd_bboxes = (const float*)d_in[1];
    const float* anc       = (const float*)d_in[2];
    const int*   gt_labels = (const int*)d_in[3];
    const float* gt_bboxes = (const float*)d_in[4];

    int grids = in_sizes[2] / 2;                                   // anc_points: (grids, 2)
    int tn    = in_sizes[3];                                       // gt_labels: (tn, 1)
    int nc    = (int)((long long)in_sizes[0] / ((long long)tn * grids));

    size_t N = (size_t)tn * grids;

    // Output layout: [labels:int32 N][bboxes:f32 4N][scores:f32 nc*N][mask:f32 N]
    float* out        = (float*)d_out;
    int*   out_labels = (int*)out;
    float* out_bboxes = out + N;
    float* out_scores = out + N * 5;
    float* out_mask   = out + N * (size_t)(5 + nc);

    // Workspace layout
    float* wsf      = (float*)d_ws;
    float* metric   = wsf;                       // N floats
    float* ovl_enc  = wsf + N;                   // N floats
    float* rowscale = wsf + 2 * N;               // tn floats
    int*   pos_idx  = (int*)(rowscale + tn);     // tn*TOPK ints
    float* pos_am   = (float*)(pos_idx + (size_t)tn * TOPK);

    // K0: zero the whole output
    long long n  = (long long)out_size;
    long long n4 = n / 4;
    long long zb = (n4 + 255) / 256;
    zero_out_kernel<<<(unsigned)zb, 256, 0, stream>>>(out, n4, n);

    // K1: metrics
    dim3 mgrid((grids + 255) / 256, tn);
    metrics_kernel<<<mgrid, 256, 0, stream>>>(pd_scores, pd_bboxes, anc,
                                              gt_labels, gt_bboxes,
                                              metric, ovl_enc, grids, nc);

    // K2: per-row top-13 (async LDS staging)
    topk_kernel<<<tn, 256, 0, stream>>>(metric, ovl_enc,
                                        pos_idx, pos_am, rowscale, grids);

    // K3: scatter positives
    int np = tn * TOPK;
    scatter_kernel<<<(np + 255) / 256, 256, 0, stream>>>(gt_labels, gt_bboxes,
                                                         pos_idx, pos_am, rowscale,
                                                         out_labels, out_bboxes,
                                                         out_scores, out_mask,
                                                         grids, nc, tn);
}